// CapsNetBCL_44985487458960
// MI455X (gfx1250) — compile-verified
//
#include <hip/hip_runtime.h>
#include <hip/hip_bf16.h>

// ---------------------------------------------------------------------------
// CapsNet forward for MI455X (gfx1250), wave32, WMMA f32 16x16x4.
//
// Sizes: NTASKS=10, CAP=8, L=256, D=768, B=64.  rows = B*L = 16384.
// Algebraic fusion: h2 = x @ Wf^T + bf, Wf[t,c,:] = fc2_w[t,c,:] @ fc1_w[t],
// bf = fc2_w @ fc1_b + fc2_b  (no activation between fc1 and fc2).
// Final output uses pre-squash `vote` of the last routing iteration.
//
// Bandwidth plan (23.3 TB/s HBM): route_weights 168 MB + x 50 MB streamed
// once with NT hint; out 50 MB NT store; all intermediates (~16 MB) L2-resident.
// ---------------------------------------------------------------------------

typedef float v2f __attribute__((ext_vector_type(2)));
typedef float v8f __attribute__((ext_vector_type(8)));
typedef int   v4i __attribute__((vector_size(16)));   // matches async-LDS builtin pointee

typedef __attribute__((address_space(1))) v4i* global_v4i_ptr;
typedef __attribute__((address_space(3))) v4i* lds_v4i_ptr;

// D = A(16x4) * B(4x16) + C, all fp32, exact.
// Fragment layout (ISA 7.12.2): lanes 0-15 hold K={0,1}, lanes 16-31 K={2,3}.
__device__ __forceinline__ v8f wmma_f32(v2f a, v2f b, v8f c) {
    return __builtin_amdgcn_wmma_f32_16x16x4_f32(
        /*neg_a=*/false, a, /*neg_b=*/false, b,
        /*c_mod=*/(short)0, c, /*reuse_a=*/false, /*reuse_b=*/false);
}

#define NTASKS 10
#define CAP    8
#define SEQL   256
#define DIM    768
#define BATCH  64
#define ROWS   (BATCH * SEQL)   // 16384
#define NTC    (NTASKS * CAP)   // 80

#if defined(__HIP_DEVICE_COMPILE__) && \
    __has_builtin(__builtin_amdgcn_global_load_async_to_lds_b128) && \
    __has_builtin(__builtin_amdgcn_s_wait_asynccnt)
#define HAVE_ASYNC_LDS 1
#else
#define HAVE_ASYNC_LDS 0
#endif

// ---------------------------------------------------------------------------
// Kernel A: Wf[t*8+c][i] = sum_d fc2_w[t,c,d] * fc1_w[t,d,i]   (80 x 768)
// ---------------------------------------------------------------------------
__global__ void __launch_bounds__(256) fuse_weights(
    const float* __restrict__ fc1w, const float* __restrict__ fc2w,
    float* __restrict__ Wf) {
    int t = blockIdx.x / 3;
    int i = (blockIdx.x % 3) * 256 + threadIdx.x;
    const float* f1 = fc1w + (size_t)t * DIM * DIM + i;
    const float* f2 = fc2w + (size_t)t * CAP * DIM;
    float acc[CAP] = {};
    for (int d = 0; d < DIM; ++d) {
        float v = f1[(size_t)d * DIM];
        #pragma unroll
        for (int c = 0; c < CAP; ++c) acc[c] += f2[c * DIM + d] * v;
    }
    #pragma unroll
    for (int c = 0; c < CAP; ++c)
        Wf[(size_t)(t * CAP + c) * DIM + i] = acc[c];
}

// bf[t*8+c] = fc2_b[t,c] + sum_d fc2_w[t,c,d] * fc1_b[t,d]
__global__ void fuse_bias(const float* __restrict__ fc1b,
                          const float* __restrict__ fc2w,
                          const float* __restrict__ fc2b,
                          float* __restrict__ bf) {
    int tc = blockIdx.x * blockDim.x + threadIdx.x;
    if (tc >= NTC) return;
    int t = tc >> 3;
    const float* w = fc2w + (size_t)tc * DIM;
    const float* b = fc1b + (size_t)t * DIM;
    float s = fc2b[tc];
    for (int d = 0; d < DIM; ++d) s += w[d] * b[d];
    bf[tc] = s;
}

// ---------------------------------------------------------------------------
// Kernel B: h2[row][tc] = x[row,:] . Wf[tc,:] + bf[tc]
// WMMA GEMM  M=16384, N=80 (5 tiles), K=768.  x streamed once (NT load).
// ---------------------------------------------------------------------------
__global__ void __launch_bounds__(256) fused_fc_gemm(
    const float* __restrict__ x, const float* __restrict__ Wf,
    const float* __restrict__ bf, float* __restrict__ h2) {
    int tid  = threadIdx.x;
    int lane = tid & 31;
    int w    = tid >> 5;
    int row0 = (blockIdx.x * 8 + w) * 16;
    int half = lane >> 4;
    int mr   = lane & 15;

    v8f acc[5] = {};
    const float* arow = x + (size_t)(row0 + mr) * DIM + 2 * half;
    #pragma unroll 2
    for (int k = 0; k < DIM; k += 4) {
        v2f a = __builtin_nontemporal_load((const v2f*)(arow + k));
        #pragma unroll
        for (int j = 0; j < 5; ++j) {
            // B[k][n] = Wf[n][k] : K-contiguous in Wf row -> b64 load (L2 hit)
            v2f b = *(const v2f*)(Wf + (size_t)(j * 16 + mr) * DIM + k + 2 * half);
            acc[j] = wmma_f32(a, b, acc[j]);
        }
    }
    #pragma unroll
    for (int j = 0; j < 5; ++j) {
        int col = j * 16 + mr;
        float bias = bf[col];
        #pragma unroll
        for (int r = 0; r < 8; ++r)
            h2[(size_t)(row0 + r + 8 * half) * NTC + col] = acc[j][r] + bias;
    }
}

// ---------------------------------------------------------------------------
// Kernel B2: squash over task dim, emit sem[b][t][l*8+c].
// ---------------------------------------------------------------------------
__global__ void __launch_bounds__(256) squash_sem(
    const float* __restrict__ h2, float* __restrict__ sem) {
    int g   = blockIdx.x * 256 + threadIdx.x;  // 131072 = ROWS*CAP
    int row = g >> 3;
    int c   = g & 7;
    int b   = row >> 8;
    int l   = row & 255;
    const float* hp = h2 + (size_t)row * NTC + c;
    float v[NTASKS];
    float sq = 1e-16f;
    #pragma unroll
    for (int t = 0; t < NTASKS; ++t) { v[t] = hp[t * CAP]; sq += v[t] * v[t]; }
    float scale = (sq / (1.f + sq)) * rsqrtf(sq);
    size_t ob = (size_t)b * NTASKS * 2048 + (size_t)l * CAP + c;
    #pragma unroll
    for (int t = 0; t < NTASKS; ++t) sem[ob + (size_t)t * 2048] = v[t] * scale;
}

// ---------------------------------------------------------------------------
// Kernel C: priors[c][b][t][o] = sum_i sem[b][t][i] * rw[c][t][i][o]
// 80 batched WMMA GEMMs, M=64, K=2048, N=256. Streams rw (168 MB) once.
// grid (80,4), block 256: wave -> (m-tile 0..3, 32-col group).
// Async path: 16x64 rw slab DMA'd to LDS (global_load_async_to_lds_b128),
// double-buffered; copy of slab s+1 overlaps 8 WMMAs of slab s.  LDS row
// stride 80 floats -> the two 16-lane halves hit disjoint bank groups.
// ---------------------------------------------------------------------------
__global__ void __launch_bounds__(256) priors_gemm(
    const float* __restrict__ sem, const float* __restrict__ rw,
    float* __restrict__ priors) {
    int ct   = blockIdx.x;          // c*10 + t  (matches rw [CAP][NTASKS] order)
    int c    = ct / NTASKS;
    int t    = ct % NTASKS;
    int tid  = threadIdx.x;
    int lane = tid & 31;
    int w    = tid >> 5;
    int m0   = (w & 3) * 16;        // m-tile
    int lc0  = (w >> 2) * 32;       // local col base (2 n-tiles of 16)
    int o0b  = blockIdx.y * 64;     // block's col window
    int half = lane >> 4;
    int mr   = lane & 15;

    const float* A  = sem + ((size_t)(m0 + mr) * NTASKS + t) * 2048 + 2 * half;
    const float* RW = rw + (size_t)ct * 2048 * 256 + o0b;

    v8f acc0 = {}, acc1 = {};

#if HAVE_ASYNC_LDS
    __shared__ __align__(16) float lbuf[2][16 * 80];   // padded stride 80
    int sr = tid >> 4;              // staged row 0..15
    int sc = (tid & 15) * 4;        // staged col (floats), 16B chunk
    auto stage = [&](int buf, int k0) {
        __builtin_amdgcn_global_load_async_to_lds_b128(
            (global_v4i_ptr)(RW + (size_t)(k0 + sr) * 256 + sc),
            (lds_v4i_ptr)(&lbuf[buf][sr * 80 + sc]),
            0, 0);
    };
    stage(0, 0);
    for (int s = 0; s < 128; ++s) {          // 128 slabs of K=16
        int buf = s & 1;
        __builtin_amdgcn_s_wait_asynccnt(0);
        __syncthreads();                     // slab `buf` visible to all waves
        if (s + 1 < 128) stage(buf ^ 1, (s + 1) * 16);
        const float* L = &lbuf[buf][0];
        #pragma unroll
        for (int kk = 0; kk < 16; kk += 4) {
            v2f a = *(const v2f*)(A + s * 16 + kk);
            int r0 = (kk + 2 * half) * 80;
            v2f b0; b0.x = L[r0 + lc0 + mr];      b0.y = L[r0 + 80 + lc0 + mr];
            v2f b1; b1.x = L[r0 + lc0 + 16 + mr]; b1.y = L[r0 + 80 + lc0 + 16 + mr];
            acc0 = wmma_f32(a, b0, acc0);
            acc1 = wmma_f32(a, b1, acc1);
        }
        __syncthreads();                     // done reading `buf` before s+1 stages it
    }
#else
    const float* B0 = RW + (size_t)(2 * half) * 256 + lc0 + mr;
    #pragma unroll 4
    for (int k = 0; k < 2048; k += 4) {
        v2f a = *(const v2f*)(A + k);
        const float* bp = B0 + (size_t)k * 256;
        v2f b0; b0.x = __builtin_nontemporal_load(bp);
                b0.y = __builtin_nontemporal_load(bp + 256);
        v2f b1; b1.x = __builtin_nontemporal_load(bp + 16);
                b1.y = __builtin_nontemporal_load(bp + 256 + 16);
        acc0 = wmma_f32(a, b0, acc0);
        acc1 = wmma_f32(a, b1, acc1);
    }
#endif

    #pragma unroll
    for (int r = 0; r < 8; ++r) {
        int bb = m0 + r + 8 * half;
        size_t base = (((size_t)c * BATCH + bb) * NTASKS + t) * 256 + o0b + lc0 + mr;
        priors[base]      = acc0[r];
        priors[base + 16] = acc1[r];
    }
}

// ---------------------------------------------------------------------------
// Kernel D: dynamic routing (3 iters). One block (256 thr) per (c,b).
// logits are uniform over o (init 0, per-t scalar updates) -> keep per-t.
// Output = vote of last iteration (reference uses pre-squash vote).
// ---------------------------------------------------------------------------
__device__ __forceinline__ float block_reduce(float v, float* red) {
    #pragma unroll
    for (int off = 16; off > 0; off >>= 1) v += __shfl_xor(v, off, 32);
    int lane = threadIdx.x & 31, w = threadIdx.x >> 5;
    if (lane == 0) red[w] = v;
    __syncthreads();
    float r = 0.f;
    #pragma unroll
    for (int i = 0; i < 8; ++i) r += red[i];
    __syncthreads();
    return r;
}

__global__ void __launch_bounds__(256) routing(
    const float* __restrict__ priors, const int* __restrict__ evalp,
    float* __restrict__ vote_out) {
    __shared__ float red[8];
    int cb = blockIdx.x;        // c*64 + b
    int o  = threadIdx.x;
    int et = *evalp;

    float p[NTASKS], lg[NTASKS];
    const float* pb = priors + (size_t)cb * NTASKS * 256 + o;
    #pragma unroll
    for (int t = 0; t < NTASKS; ++t) p[t] = pb[t * 256];
    #pragma unroll
    for (int t = 0; t < NTASKS; ++t) lg[t] = (t <= et) ? 0.f : -10000.f;

    float vote = 0.f;
    for (int it = 0; it < 3; ++it) {
        float mx = lg[0];
        #pragma unroll
        for (int t = 1; t < NTASKS; ++t) mx = fmaxf(mx, lg[t]);
        float e[NTASKS], s = 0.f;
        #pragma unroll
        for (int t = 0; t < NTASKS; ++t) { e[t] = __expf(lg[t] - mx); s += e[t]; }
        float inv = 1.f / s;
        vote = 0.f;
        #pragma unroll
        for (int t = 0; t < NTASKS; ++t) vote += e[t] * inv * p[t];
        if (it == 2) break;
        float sq = block_reduce(vote * vote, red) + 1e-16f;
        float out = vote * (sq / (1.f + sq)) * rsqrtf(sq);
        for (int t = 0; t < NTASKS; ++t) {
            float d = block_reduce(p[t] * out, red);
            lg[t] = (t <= et) ? (lg[t] + d) : -10000.f;
        }
    }
    vote_out[(size_t)cb * 256 + o] = vote;   // flat [CAP][B][L] == flat [B][L][CAP] view
}

// ---------------------------------------------------------------------------
// Kernel E: out[row][d] = sum_c vote[row*8+c] * lw[et][d][c] + lb[et][d]
// WMMA GEMM  M=16384, N=768, K=8 (2 K-steps); NT stores (write-once output).
// ---------------------------------------------------------------------------
__global__ void __launch_bounds__(256) out_gemm(
    const float* __restrict__ vote, const float* __restrict__ lw,
    const float* __restrict__ lb, const int* __restrict__ evalp,
    float* __restrict__ out) {
    int et = *evalp;
    const float* W  = lw + (size_t)et * DIM * CAP;
    const float* Bv = lb + (size_t)et * DIM;
    int tid  = threadIdx.x;
    int lane = tid & 31;
    int w    = tid >> 5;
    int row0 = blockIdx.x * 16;
    int half = lane >> 4;
    int mr   = lane & 15;

    const float* ap = vote + (size_t)(row0 + mr) * CAP + 2 * half;
    v2f a0 = *(const v2f*)(ap);
    v2f a1 = *(const v2f*)(ap + 4);

    #pragma unroll
    for (int j = 0; j < 6; ++j) {
        int col = (w * 6 + j) * 16 + mr;
        const float* bp = W + (size_t)col * CAP + 2 * half;
        v2f b0 = *(const v2f*)(bp);
        v2f b1 = *(const v2f*)(bp + 4);
        v8f acc = {};
        acc = wmma_f32(a0, b0, acc);
        acc = wmma_f32(a1, b1, acc);
        float bias = Bv[col];
        #pragma unroll
        for (int r = 0; r < 8; ++r)
            __builtin_nontemporal_store(
                acc[r] + bias, out + (size_t)(row0 + r + 8 * half) * DIM + col);
    }
}

// ---------------------------------------------------------------------------
extern "C" void kernel_launch(void* const* d_in, const int* in_sizes, int n_in,
                              void* d_out, int out_size, void* d_ws, size_t ws_size,
                              hipStream_t stream) {
    const float* x    = (const float*)d_in[0];
    const float* fc1w = (const float*)d_in[1];
    const float* fc1b = (const float*)d_in[2];
    const float* fc2w = (const float*)d_in[3];
    const float* fc2b = (const float*)d_in[4];
    const float* rw   = (const float*)d_in[5];
    const float* lw   = (const float*)d_in[6];
    const float* lb   = (const float*)d_in[7];
    const int*   evp  = (const int*)d_in[8];

    float* ws   = (float*)d_ws;
    float* Wf   = ws;                 //    80*768   =    61,440 f
    float* bf   = ws + 65536;         //    80 f
    float* h2   = ws + 131072;        // 16384*80    = 1,310,720 f
    float* sem  = ws + 1441792;       // 64*10*2048  = 1,310,720 f
    float* pri  = ws + 2752512;       // 8*64*10*256 = 1,310,720 f
    float* vote = ws + 4063232;       // 8*64*256    =   131,072 f  (end 16 MiB)
    float* out  = (float*)d_out;

    fuse_weights <<<30,   256, 0, stream>>>(fc1w, fc2w, Wf);
    fuse_bias    <<<1,    128, 0, stream>>>(fc1b, fc2w, fc2b, bf);
    fused_fc_gemm<<<128,  256, 0, stream>>>(x, Wf, bf, h2);
    squash_sem   <<<512,  256, 0, stream>>>(h2, sem);
    dim3 gp(80, 4);
    priors_gemm  <<<gp,   256, 0, stream>>>(sem, rw, pri);
    routing      <<<512,  256, 0, stream>>>(pri, evp, vote);
    out_gemm     <<<1024, 256, 0, stream>>>(vote, lw, lb, evp, out);
    (void)in_sizes; (void)n_in; (void)out_size; (void)ws_size;
}